// QRNN_5385888989480
// MI455X (gfx1250) — compile-verified
//
#include <hip/hip_runtime.h>

// ---------------------------------------------------------------------------
// QRNN (window=2 causal conv -> fo-pool) for MI455X / gfx1250, wave32.
//
// Phase 1: gates GEMM  [B*T, 1024] x [1024, 1536]  via V_WMMA_F32_16X16X4_F32
//          with double-buffered GLOBAL_LOAD_ASYNC_TO_LDS_B128 staging
//          (register-staged b128 fallback if the async builtin is absent).
// Phase 2: chunked parallel linear scan (c_t = f*c + (1-f)*z), y = o*c
// ---------------------------------------------------------------------------

typedef float v2f __attribute__((ext_vector_type(2)));
typedef float v8f __attribute__((ext_vector_type(8)));
typedef int   v4i __attribute__((ext_vector_type(4)));

#define B_      8
#define T_      4096
#define D_      512
#define U_      512
#define N_      1536          // 3*U
#define K_      1024          // 2*D
#define M_      (B_*T_)       // 32768
#define CHUNK   128
#define NCHUNK  (T_/CHUNK)    // 32
#define KC      32            // K staging chunk (divides 512 -> tap uniform)
#define A_STRIDE 36           // 32 + 4 pad (16B aligned, conflict-free col reads)
#define B_STRIDE 144          // 128 + 16 pad (mod 64 = 16: halves hit disjoint banks)

#if defined(__gfx1250__) && \
    __has_builtin(__builtin_amdgcn_global_load_async_to_lds_b128) && \
    __has_builtin(__builtin_amdgcn_s_wait_asynccnt)
#define USE_ASYNC_LDS 1
#else
#define USE_ASYNC_LDS 0
#endif

#if USE_ASYNC_LDS
// Builtin signature (from clang diagnostics): param0 = int4 in AS(1) (global),
// param1 = LDS side, params 2/3 = constant offset / cpol.
typedef __attribute__((address_space(1))) v4i g_v4i;
typedef __attribute__((address_space(3))) v4i l_v4i;
__device__ __forceinline__ g_v4i* as_glob(const void* p) {
    return (g_v4i*)(uintptr_t)p;             // flat global addr value == AS1 addr
}
__device__ __forceinline__ l_v4i* as_lds(void* p) {
    return (l_v4i*)(uintptr_t)p;             // flat LDS addr low bits == LDS offset
}
#endif

__device__ __forceinline__ float fast_sigmoid(float v) {
    return 1.0f / (1.0f + __expf(-v));
}
__device__ __forceinline__ float fast_tanh(float v) {
    return 2.0f / (1.0f + __expf(-2.0f * v)) - 1.0f;   // one v_exp_f32
}

// ---------------------------------------------------------------------------
// K1: gates GEMM + bias + activations -> z, f, o  (each [B,T,U] f32)
// grid = (N_/128, M_/64), block = 256 (8 waves).
// Wave layout in 64x128 block tile: 4 waves down M, 2 across N; each wave
// owns a 16(M) x 64(N) strip = 4 accumulators of 16x16.
// ---------------------------------------------------------------------------
__global__ __launch_bounds__(256)
void qrnn_gates_kernel(const float* __restrict__ x,
                       const float* __restrict__ w,      // [1024,1536] row-major
                       const float* __restrict__ bias,   // [1536]
                       float* __restrict__ z,
                       float* __restrict__ f,
                       float* __restrict__ o)
{
    __shared__ float Alds[2][64 * A_STRIDE];
    __shared__ float Blds[2][KC * B_STRIDE];

    const int tid  = threadIdx.x;
    const int wave = tid >> 5;
    const int lane = tid & 31;
    const int half = lane >> 4;            // 0: lanes 0-15, 1: lanes 16-31
    const int ln   = lane & 15;

    const int n_blk = blockIdx.x * 128;
    const int m_blk = blockIdx.y * 64;

    const int wm = (wave & 3) * 16;        // wave M offset in block tile
    const int wn = (wave >> 2) * 64;       // wave N offset in block tile

    // ---- A staging geometry: 64 rows x 32 k = 512 float4; 2 per thread ----
    const int aRow0 = tid >> 3;            // 0..31 (second row = +32)
    const int aK4   = (tid & 7) * 4;       // column (k within chunk)
    const int r0    = m_blk + aRow0;
    const int r1    = r0 + 32;
    const int b0 = r0 >> 12, t0 = r0 & (T_ - 1);
    const int b1 = r1 >> 12, t1 = r1 & (T_ - 1);   // t1 >= 32 always
    const size_t rowbase0 = (size_t)(b0 * T_ + t0) * D_;
    const size_t rowbase1 = (size_t)(b1 * T_ + t1) * D_;
    const int aOff0 = aRow0 * A_STRIDE + aK4;
    const int aOff1 = (aRow0 + 32) * A_STRIDE + aK4;

    // ---- B staging geometry: 32 k x 128 n = 1024 float4; 4 per thread ----
    // wave v stages rows {v, v+8, v+16, v+24}; each row = 32 lanes x float4.
    const int bN = lane * 4;

    v8f acc[4];
    #pragma unroll
    for (int i = 0; i < 4; ++i) acc[i] = (v8f)0.0f;

#if USE_ASYNC_LDS
    // 5 or 6 async b128 copies per lane per chunk (2 A + 4 B; the causal-pad
    // row is a plain LDS zero store), in-order completion.
    auto stage = [&](int bi, int kc) {
        float* Ab = &Alds[bi][0];
        float* Bb = &Blds[bi][0];
        if (kc < D_) {                     // tap 0: x[b, t-1, kc..]
            if (t0 == 0) {                 // causal zero pad (plain LDS store)
                float4 z4 = make_float4(0.f, 0.f, 0.f, 0.f);
                *(float4*)(Ab + aOff0) = z4;
            } else {
                __builtin_amdgcn_global_load_async_to_lds_b128(
                    as_glob(x + rowbase0 - D_ + kc + aK4), as_lds(Ab + aOff0), 0, 0);
            }
            __builtin_amdgcn_global_load_async_to_lds_b128(
                as_glob(x + rowbase1 - D_ + kc + aK4), as_lds(Ab + aOff1), 0, 0);
        } else {                           // tap 1: x[b, t, kc-512..]
            __builtin_amdgcn_global_load_async_to_lds_b128(
                as_glob(x + rowbase0 + (kc - D_) + aK4), as_lds(Ab + aOff0), 0, 0);
            __builtin_amdgcn_global_load_async_to_lds_b128(
                as_glob(x + rowbase1 + (kc - D_) + aK4), as_lds(Ab + aOff1), 0, 0);
        }
        #pragma unroll
        for (int i = 0; i < 4; ++i) {
            const int kk = i * 8 + wave;
            __builtin_amdgcn_global_load_async_to_lds_b128(
                as_glob(w + (size_t)(kc + kk) * N_ + n_blk + bN),
                as_lds(Bb + kk * B_STRIDE + bN), 0, 0);
        }
    };
#else
    float4 aR[2], bR[4];
    auto load_regs = [&](int kc) {         // issue all 6 b128 loads, no waits
        if (kc < D_) {
            aR[0] = (t0 == 0) ? make_float4(0.f, 0.f, 0.f, 0.f)
                              : *(const float4*)(x + rowbase0 - D_ + kc + aK4);
            aR[1] = *(const float4*)(x + rowbase1 - D_ + kc + aK4);
        } else {
            aR[0] = *(const float4*)(x + rowbase0 + (kc - D_) + aK4);
            aR[1] = *(const float4*)(x + rowbase1 + (kc - D_) + aK4);
        }
        #pragma unroll
        for (int i = 0; i < 4; ++i) {
            const int kk = i * 8 + wave;
            bR[i] = *(const float4*)(w + (size_t)(kc + kk) * N_ + n_blk + bN);
        }
    };
    auto store_regs = [&](int bi) {
        *(float4*)(&Alds[bi][aOff0]) = aR[0];
        *(float4*)(&Alds[bi][aOff1]) = aR[1];
        #pragma unroll
        for (int i = 0; i < 4; ++i) {
            const int kk = i * 8 + wave;
            *(float4*)(&Blds[bi][kk * B_STRIDE + bN]) = bR[i];
        }
    };
#endif

    auto compute = [&](int bi) {
        const float* Ab = &Alds[bi][0];
        const float* Bb = &Blds[bi][0];
        #pragma unroll
        for (int k4 = 0; k4 < KC; k4 += 4) {
            // A 16x4 f32: low lanes K0(v0),K1(v1); high lanes K2,K3
            const int kA = k4 + 2 * half;
            v2f a;
            a.x = Ab[(wm + ln) * A_STRIDE + kA];
            a.y = Ab[(wm + ln) * A_STRIDE + kA + 1];
            #pragma unroll
            for (int nt = 0; nt < 4; ++nt) {
                // B 4x16 f32: rows striped over lanes; low K0/K1, high K2/K3
                v2f bf;
                bf.x = Bb[kA * B_STRIDE + wn + nt * 16 + ln];
                bf.y = Bb[(kA + 1) * B_STRIDE + wn + nt * 16 + ln];
                acc[nt] = __builtin_amdgcn_wmma_f32_16x16x4_f32(
                    false, a, false, bf, (short)0, acc[nt], false, false);
            }
        }
    };

#if USE_ASYNC_LDS
    // Double-buffered async pipeline: issue next chunk, retire current, barrier,
    // compute, barrier (frees buffer for the following issue).
    // Wait threshold must be <= min copies issued per chunk (5, for causal-pad
    // threads): current c in {5,6}, next n in {5,6} outstanding ->
    // wait<=5 retires c+n-5 >= c, i.e. the whole current chunk, always.
    stage(0, 0);
    int cur = 0;
    for (int kc = 0; kc < K_; kc += KC) {
        const int next = kc + KC;
        if (next < K_) {
            stage(cur ^ 1, next);
            __builtin_amdgcn_s_wait_asynccnt(5);
        } else {
            __builtin_amdgcn_s_wait_asynccnt(0);
        }
        __syncthreads();
        compute(cur);
        __syncthreads();
        cur ^= 1;
    }
#else
    // Register-staged pipeline: next chunk's global loads overlap compute.
    load_regs(0);
    int cur = 0;
    for (int kc = 0; kc < K_; kc += KC) {
        store_regs(cur);
        __syncthreads();
        if (kc + KC < K_) load_regs(kc + KC);
        compute(cur);
        cur ^= 1;                          // double buffer: no trailing barrier
    }
#endif

    // ---- epilogue: bias + activation, scatter into z / f / o ----
    // C/D layout: acc[nt][i] -> row (wm + i + 8*half), col (wn + nt*16 + ln)
    #pragma unroll
    for (int nt = 0; nt < 4; ++nt) {
        const int n_glob = n_blk + wn + nt * 16 + ln;
        const float bv   = bias[n_glob];
        const int which  = n_glob >> 9;        // /512: 0=z, 1=f, 2=o
        const int u      = n_glob & (U_ - 1);
        #pragma unroll
        for (int i = 0; i < 8; ++i) {
            int r = m_blk + wm + i + 8 * half;
            int b = r >> 12;
            int t = r & (T_ - 1);
            float g = acc[nt][i] + bv;
            size_t idx = (size_t)(b * T_ + t) * U_ + u;
            if (which == 0)      z[idx] = fast_tanh(g);
            else if (which == 1) f[idx] = fast_sigmoid(g);
            else                 o[idx] = fast_sigmoid(g);
        }
    }
}

// ---------------------------------------------------------------------------
// K2: per-chunk affine composition.  c_out = P*c_in + Q over CHUNK steps.
// One thread per (b, chunk, u); u fastest -> fully coalesced.
// ---------------------------------------------------------------------------
__global__ __launch_bounds__(256)
void qrnn_chunk_reduce_kernel(const float* __restrict__ z,
                              const float* __restrict__ f,
                              float* __restrict__ P,
                              float* __restrict__ Q)
{
    int idx = blockIdx.x * blockDim.x + threadIdx.x;   // B*NCHUNK*U total
    int u   = idx & (U_ - 1);
    int ch  = (idx >> 9) & (NCHUNK - 1);
    int b   = idx >> 14;                               // /(U_*NCHUNK)

    size_t base = (size_t)(b * T_ + ch * CHUNK) * U_ + u;
    float Pv = 1.0f, Qv = 0.0f;
    for (int t = 0; t < CHUNK; ++t) {
        float ft = f[base + (size_t)t * U_];
        float zt = z[base + (size_t)t * U_];
        Qv = fmaf(ft, Qv, (1.0f - ft) * zt);
        Pv *= ft;
    }
    P[idx] = Pv;
    Q[idx] = Qv;
}

// ---------------------------------------------------------------------------
// K3: fold chunk prefixes (NCHUNK=32, L2-resident) for the initial c, then
// replay the chunk emitting y = o * c.
// ---------------------------------------------------------------------------
__global__ __launch_bounds__(256)
void qrnn_scan_out_kernel(const float* __restrict__ z,
                          const float* __restrict__ f,
                          const float* __restrict__ o,
                          const float* __restrict__ P,
                          const float* __restrict__ Q,
                          float* __restrict__ y)
{
    int idx = blockIdx.x * blockDim.x + threadIdx.x;
    int u   = idx & (U_ - 1);
    int ch  = (idx >> 9) & (NCHUNK - 1);
    int b   = idx >> 14;

    float c = 0.0f;
    for (int j = 0; j < ch; ++j) {        // ch uniform within a block
        int pj = (b * NCHUNK + j) * U_ + u;
        c = fmaf(P[pj], c, Q[pj]);
    }

    size_t base = (size_t)(b * T_ + ch * CHUNK) * U_ + u;
    for (int t = 0; t < CHUNK; ++t) {
        size_t k = base + (size_t)t * U_;
        float ft = f[k];
        float zt = z[k];
        c = fmaf(ft, c, (1.0f - ft) * zt);
        y[k] = o[k] * c;
    }
}

// ---------------------------------------------------------------------------
extern "C" void kernel_launch(void* const* d_in, const int* in_sizes, int n_in,
                              void* d_out, int out_size, void* d_ws, size_t ws_size,
                              hipStream_t stream)
{
    const float* x    = (const float*)d_in[0];   // [B,T,D]
    const float* w    = (const float*)d_in[1];   // [2,D,3U] == [1024,1536] flat
    const float* bias = (const float*)d_in[2];   // [3U]
    float* y = (float*)d_out;                    // [B,T,U]

    const size_t BTU = (size_t)B_ * T_ * U_;     // 16,777,216 floats
    float* z = (float*)d_ws;
    float* f = z + BTU;
    float* o = f + BTU;
    float* P = o + BTU;
    float* Q = P + (size_t)B_ * NCHUNK * U_;

    // K1: gates GEMM (M=32768, N=1536, K=1024), WMMA f32
    dim3 g1(N_ / 128, M_ / 64);                  // (12, 512)
    qrnn_gates_kernel<<<g1, 256, 0, stream>>>(x, w, bias, z, f, o);

    // K2 + K3: chunked linear scan, 131072 threads each
    int scan_threads = B_ * NCHUNK * U_;
    qrnn_chunk_reduce_kernel<<<scan_threads / 256, 256, 0, stream>>>(z, f, P, Q);
    qrnn_scan_out_kernel<<<scan_threads / 256, 256, 0, stream>>>(z, f, o, P, Q, y);
}